// WeightedKappaLoss_51677046505515
// MI455X (gfx1250) — compile-verified
//
#include <hip/hip_runtime.h>

#define NC   6
#define BINS 36          // NC * NC
#define WAVES_PER_BLOCK 8
#define THREADS 256

typedef __attribute__((ext_vector_type(2))) float v2f;
typedef __attribute__((ext_vector_type(8))) float v8f;

// quadratic weight for flat bin q (row i, col j): (i-j)^2
__device__ __forceinline__ float wq(int q) {
    int i = q / NC, j = q % NC;
    float d = (float)(i - j);
    return d * d;
}

// ---------------------------------------------------------------------------
// Phase 1: streaming confusion-count histogram.
// Each 256-thread block keeps 8 per-wave LDS histograms (ds_add_u32, no
// inter-wave contention), then writes one 36-bin partial per block to d_ws.
// No global atomics; d_ws fully overwritten every call (deterministic).
// ---------------------------------------------------------------------------
__global__ __launch_bounds__(THREADS) void
confusion_kernel(const float* __restrict__ y_pred,
                 const int*   __restrict__ y_true,
                 unsigned int* __restrict__ parts,
                 int N)
{
    __shared__ unsigned int hist[WAVES_PER_BLOCK][BINS];

    const int wave = threadIdx.x >> 5;

    for (int b = threadIdx.x; b < WAVES_PER_BLOCK * BINS; b += THREADS)
        ((unsigned int*)hist)[b] = 0u;
    __syncthreads();

    const int stride = gridDim.x * THREADS;
    for (int i = blockIdx.x * THREADS + threadIdx.x; i < N; i += stride) {
        const float* row = y_pred + (size_t)i * NC;

        // prefetch next iteration's row (global_prefetch_b8)
        int ni = i + stride;
        if (ni < N)
            __builtin_prefetch(y_pred + (size_t)ni * NC, 0, 1);

        // rows are 24B apart -> 8B aligned: three b64 loads
        v2f r0 = *(const v2f*)(row);
        v2f r1 = *(const v2f*)(row + 2);
        v2f r2 = *(const v2f*)(row + 4);

        // argmax with first-occurrence tie-break (strict >)
        float m = r0.x; int p = 0;
        if (r0.y > m) { m = r0.y; p = 1; }
        if (r1.x > m) { m = r1.x; p = 2; }
        if (r1.y > m) { m = r1.y; p = 3; }
        if (r2.x > m) { m = r2.x; p = 4; }
        if (r2.y > m) { m = r2.y; p = 5; }

        int t = y_true[i];
        atomicAdd(&hist[wave][t * NC + p], 1u);   // ds_add_u32
    }
    __syncthreads();

    for (int b = threadIdx.x; b < BINS; b += THREADS) {
        unsigned int s = 0;
        #pragma unroll
        for (int w = 0; w < WAVES_PER_BLOCK; ++w) s += hist[w][b];
        parts[(size_t)blockIdx.x * BINS + b] = s;
    }
}

// ---------------------------------------------------------------------------
// Phase 2: reduce partials, compute kappa. One block of 64 threads:
// threads 0..35 each sum one bin over all partials into LDS; wave 0 then
// computes both weighted 36-element sums with V_WMMA_F32_16X16X4_F32
// (B = all-ones, so D[m][n] = sum_k A[m][k]; column sum over m == S).
// ---------------------------------------------------------------------------
__global__ __launch_bounds__(64) void
kappa_finalize(const unsigned int* __restrict__ parts, int nparts,
               float* __restrict__ out)
{
    __shared__ float cnt_s[BINS];

    const int tid = threadIdx.x;
    if (tid < BINS) {
        unsigned int s = 0;
        for (int p = 0; p < nparts; ++p) s += parts[(size_t)p * BINS + tid];
        cnt_s[tid] = (float)s;
    }
    __syncthreads();

    if (tid < 32) {           // wave 0, EXEC all ones inside this branch
        const int lane = tid;

        float total = 0.0f;
        #pragma unroll
        for (int q = 0; q < BINS; ++q) total += cnt_s[q];

        float ht[NC], hp[NC];
        #pragma unroll
        for (int i = 0; i < NC; ++i) { ht[i] = 0.0f; hp[i] = 0.0f; }
        #pragma unroll
        for (int i = 0; i < NC; ++i)
            #pragma unroll
            for (int j = 0; j < NC; ++j) {
                float c = cnt_s[i * NC + j];
                ht[i] += c;
                hp[j] += c;
            }

        // A-matrix 16x4 f32 layout: lane m=lane%16; lanes<16 -> K=0,1 ;
        // lanes>=16 -> K=2,3 (VGPR0=even K, VGPR1=odd K).
        const int m  = lane & 15;
        const int k0 = (lane < 16) ? 0 : 2;
        const int q0 = m * 4 + k0;
        const int q1 = q0 + 1;

        v2f aC; aC.x = 0.0f; aC.y = 0.0f;   // W .* confusion counts
        v2f aE; aE.x = 0.0f; aE.y = 0.0f;   // W .* (ht outer hp)
        if (q0 < BINS) {
            float w = wq(q0);
            aC.x = w * cnt_s[q0];
            aE.x = w * ht[q0 / NC] * hp[q0 % NC];
        }
        if (q1 < BINS) {
            float w = wq(q1);
            aC.y = w * cnt_s[q1];
            aE.y = w * ht[q1 / NC] * hp[q1 % NC];
        }

        v2f bones; bones.x = 1.0f; bones.y = 1.0f;   // B == all ones
        v8f c0 = {};

        v8f d1 = __builtin_amdgcn_wmma_f32_16x16x4_f32(
            false, aC, false, bones, (short)0, c0, false, false);
        v8f d2 = __builtin_amdgcn_wmma_f32_16x16x4_f32(
            false, aE, false, bones, (short)0, c0, false, false);

        float s1 = 0.0f, s2 = 0.0f;
        #pragma unroll
        for (int r = 0; r < 8; ++r) { s1 += d1[r]; s2 += d2[r]; }
        // combine M=0..7 (lanes 0-15) with M=8..15 (lanes 16-31)
        s1 += __shfl_xor(s1, 16, 32);
        s2 += __shfl_xor(s2, 16, 32);

        // kappa = 1 - (S1/total) / (S2/total^2) = 1 - S1*total/S2
        if (lane == 0)
            out[0] = 1.0f - (s1 * total) / s2;
    }
}

extern "C" void kernel_launch(void* const* d_in, const int* in_sizes, int n_in,
                              void* d_out, int out_size, void* d_ws, size_t ws_size,
                              hipStream_t stream) {
    const float* y_pred = (const float*)d_in[0];
    const int*   y_true = (const int*)d_in[1];
    float*       out    = (float*)d_out;
    unsigned int* parts = (unsigned int*)d_ws;

    const int N = in_sizes[1];

    int nblocks = 512;
    size_t need = (size_t)nblocks * BINS * sizeof(unsigned int);
    if (need > ws_size) {
        nblocks = (int)(ws_size / (BINS * sizeof(unsigned int)));
        if (nblocks < 1) nblocks = 1;
    }

    confusion_kernel<<<nblocks, THREADS, 0, stream>>>(y_pred, y_true, parts, N);
    kappa_finalize<<<1, 64, 0, stream>>>(parts, nblocks, out);
}